// BigNet_18640158064951
// MI455X (gfx1250) — compile-verified
//
#include <hip/hip_runtime.h>

typedef __attribute__((ext_vector_type(2))) float v2f;
typedef __attribute__((ext_vector_type(8))) float v8f;

// ---------------- degree / normalization ----------------

__global__ void deg_kernel(const int* __restrict__ dst, float* __restrict__ deg, int E) {
    int e = blockIdx.x * blockDim.x + threadIdx.x;
    if (e < E) atomicAdd(&deg[dst[e]], 1.0f);
}

__global__ void dinv_kernel(const float* __restrict__ deg, float* __restrict__ dinv, int n) {
    int i = blockIdx.x * blockDim.x + threadIdx.x;
    if (i < n) dinv[i] = rsqrtf(deg[i] + 1.0f);   // +1 for self-loop; always > 0
}

// ---------------- layer 1: outer product fused with dinv scale ----------------
// p1[i][f..f+3] = dinv[i] * x[i] * W1[f..f+3]   (F_out = 256, float4-vectorized)

__global__ void layer1_p_kernel(const float* __restrict__ x, const float* __restrict__ W1,
                                const float* __restrict__ dinv, float4* __restrict__ p, int n) {
    int idx = blockIdx.x * blockDim.x + threadIdx.x;     // over n * 64 float4s
    if (idx >= n * 64) return;
    const int i  = idx >> 6;
    const int f4 = idx & 63;
    const float s = dinv[i] * x[i];
    const float4 w = ((const float4*)W1)[f4];
    p[idx] = make_float4(s * w.x, s * w.y, s * w.z, s * w.w);
}

// ---------------- fp32 WMMA GEMM:  H[n x Fout] = X[n x Fin] @ W[Fin x Fout] ----
// One wave32 computes one 16x16 output tile via K-loop of V_WMMA_F32_16X16X4_F32.
// Compile-time Fin/Fout -> all strides become immediate offsets.

template <int Fin, int Fout>
__global__ __launch_bounds__(32)
void gemm_wmma_f32_kernel(const float* __restrict__ X, const float* __restrict__ W,
                          float* __restrict__ H) {
    const int lane  = threadIdx.x;        // 0..31
    const int l15   = lane & 15;
    const int hi    = lane >> 4;          // 0: K pair {0,1}, 1: K pair {2,3}
    const int mBase = blockIdx.x * 16;
    const int nBase = blockIdx.y * 16;
    const int col   = nBase + l15;

    const float* xrow = X + (size_t)(mBase + l15) * Fin + 2 * hi;
    const float* wp   = W + (size_t)(2 * hi) * Fout + col;
    v8f c = {0.f, 0.f, 0.f, 0.f, 0.f, 0.f, 0.f, 0.f};

#pragma unroll 4
    for (int k = 0; k < Fin; k += 4) {
        v2f a;
        a.x = xrow[k];                    // contiguous pair -> b64 load
        a.y = xrow[k + 1];
        v2f b;
        b.x = wp[(size_t)k * Fout];
        b.y = wp[(size_t)(k + 1) * Fout];
        c = __builtin_amdgcn_wmma_f32_16x16x4_f32(
                /*neg_a=*/false, a, /*neg_b=*/false, b,
                /*c_mod=*/(short)0, c, /*reuse_a=*/false, /*reuse_b=*/false);
    }

    // C/D layout: VGPR v -> row M = mBase + hi*8 + v, column N = col
    float* hrow = H + (size_t)(mBase + hi * 8) * Fout + col;
#pragma unroll
    for (int v = 0; v < 8; ++v) hrow[(size_t)v * Fout] = c[v];
}

// ---------------- layer 4 GEMV: h[i] = dot(X[i, 0:64], W4) ----------------

__global__ void dot64_kernel(const float* __restrict__ X, const float* __restrict__ W4,
                             float* __restrict__ h, int n) {
    int i = blockIdx.x * blockDim.x + threadIdx.x;
    if (i >= n) return;
    const float4* r = (const float4*)(X + (size_t)i * 64);
    const float4* w = (const float4*)W4;
    float s = 0.f;
#pragma unroll
    for (int k = 0; k < 16; ++k) {
        const float4 a = r[k], b = w[k];
        s += a.x * b.x + a.y * b.y + a.z * b.z + a.w * b.w;
    }
    h[i] = s;
}

// ---------------- p = dinv * h (in place, float4) ----------------

__global__ void scale4_kernel(float4* __restrict__ p, const float* __restrict__ dinv,
                              int n, int logF4) {           // logF4 = log2(F/4)
    int idx = blockIdx.x * blockDim.x + threadIdx.x;
    if (idx >= (n << logF4)) return;
    const float s = dinv[idx >> logF4];
    float4 v = p[idx];
    v.x *= s; v.y *= s; v.z *= s; v.w *= s;
    p[idx] = v;
}

__global__ void scale1_kernel(float* __restrict__ p, const float* __restrict__ dinv, int n) {
    int i = blockIdx.x * blockDim.x + threadIdx.x;
    if (i < n) p[i] *= dinv[i];
}

// ---------------- edge scatter: acc[dst] += p[src] (float4 gather, f32 atomics) ----

__global__ void scatter4_kernel(const int* __restrict__ src, const int* __restrict__ dst,
                                const float4* __restrict__ p, float* __restrict__ acc,
                                int E, int logF4) {         // logF4 = log2(F/4)
    int idx = blockIdx.x * blockDim.x + threadIdx.x;
    if (idx >= (E << logF4)) return;
    const int e  = idx >> logF4;
    const int f4 = idx & ((1 << logF4) - 1);
    const int s  = src[e];
    const int d  = dst[e];
    const float4 v = p[(((size_t)s) << logF4) + f4];
    float* a = acc + ((((size_t)d) << logF4) + f4) * 4;
    atomicAdd(a + 0, v.x);
    atomicAdd(a + 1, v.y);
    atomicAdd(a + 2, v.z);
    atomicAdd(a + 3, v.w);
}

__global__ void scatter1_kernel(const int* __restrict__ src, const int* __restrict__ dst,
                                const float* __restrict__ p, float* __restrict__ acc, int E) {
    int e = blockIdx.x * blockDim.x + threadIdx.x;
    if (e < E) atomicAdd(&acc[dst[e]], p[src[e]]);
}

// ---------------- combine: out = act(dinv*(acc + p) + b), in place into acc ----

__global__ void combine4_kernel(float4* __restrict__ acc, const float4* __restrict__ p,
                                const float* __restrict__ dinv, const float4* __restrict__ bias,
                                int n, int logF4, int relu) {
    int idx = blockIdx.x * blockDim.x + threadIdx.x;
    if (idx >= (n << logF4)) return;
    const int i  = idx >> logF4;
    const int f4 = idx & ((1 << logF4) - 1);
    const float s = dinv[i];
    const float4 a = acc[idx], q = p[idx], b = bias[f4];
    float4 v = make_float4(s * (a.x + q.x) + b.x, s * (a.y + q.y) + b.y,
                           s * (a.z + q.z) + b.z, s * (a.w + q.w) + b.w);
    if (relu) {
        v.x = fmaxf(v.x, 0.f); v.y = fmaxf(v.y, 0.f);
        v.z = fmaxf(v.z, 0.f); v.w = fmaxf(v.w, 0.f);
    }
    acc[idx] = v;
}

// ---------------- final combine -> d_out (F=1, no relu) ----------------

__global__ void combine_out_kernel(const float* __restrict__ acc, const float* __restrict__ p,
                                   const float* __restrict__ dinv, const float* __restrict__ bias,
                                   float* __restrict__ out, int n) {
    int i = blockIdx.x * blockDim.x + threadIdx.x;
    if (i < n) out[i] = dinv[i] * (acc[i] + p[i]) + bias[0];
}

// ---------------- host driver ----------------

extern "C" void kernel_launch(void* const* d_in, const int* in_sizes, int n_in,
                              void* d_out, int out_size, void* d_ws, size_t ws_size,
                              hipStream_t stream) {
    const float* x  = (const float*)d_in[0];
    const int*   ei = (const int*)d_in[1];
    const float* W1 = (const float*)d_in[2]; const float* b1 = (const float*)d_in[3];
    const float* W2 = (const float*)d_in[4]; const float* b2 = (const float*)d_in[5];
    const float* W3 = (const float*)d_in[6]; const float* b3 = (const float*)d_in[7];
    const float* W4 = (const float*)d_in[8]; const float* b4 = (const float*)d_in[9];

    const int n = in_sizes[0];          // 50000 (divisible by 16)
    const int E = in_sizes[1] / 2;      // 800000
    const int* src = ei;
    const int* dst = ei + E;

    // workspace layout: dinv | bufA (n*256 f32) | bufB (n*256 f32)
    float* dinv = (float*)d_ws;
    float* A    = dinv + 50176;                 // 16-byte aligned past 50000
    float* B    = A + (size_t)n * 256;

    auto cdiv = [](int a, int b) { return (a + b - 1) / b; };
    const int TB = 256;

    // --- degrees / dinv (deg staged in A) ---
    hipMemsetAsync(A, 0, (size_t)n * sizeof(float), stream);
    deg_kernel<<<cdiv(E, TB), TB, 0, stream>>>(dst, A, E);
    dinv_kernel<<<cdiv(n, TB), TB, 0, stream>>>(A, dinv, n);

    // --- layer 1: Fin=1 -> Fout=256 (logF4 = 6) ---
    layer1_p_kernel<<<cdiv(n * 64, TB), TB, 0, stream>>>(x, W1, dinv, (float4*)A, n);
    hipMemsetAsync(B, 0, (size_t)n * 256 * sizeof(float), stream);
    scatter4_kernel<<<cdiv(E << 6, TB), TB, 0, stream>>>(src, dst, (const float4*)A, B, E, 6);
    combine4_kernel<<<cdiv(n << 6, TB), TB, 0, stream>>>((float4*)B, (const float4*)A, dinv,
                                                         (const float4*)b1, n, 6, 1);
    // out1 = B [n x 256]

    // --- layer 2: 256 -> 128 (WMMA fp32), logF4 = 5 ---
    {
        dim3 grid(n / 16, 128 / 16);
        gemm_wmma_f32_kernel<256, 128><<<grid, 32, 0, stream>>>(B, W2, A);
    }
    scale4_kernel<<<cdiv(n << 5, TB), TB, 0, stream>>>((float4*)A, dinv, n, 5);
    hipMemsetAsync(B, 0, (size_t)n * 128 * sizeof(float), stream);
    scatter4_kernel<<<cdiv(E << 5, TB), TB, 0, stream>>>(src, dst, (const float4*)A, B, E, 5);
    combine4_kernel<<<cdiv(n << 5, TB), TB, 0, stream>>>((float4*)B, (const float4*)A, dinv,
                                                         (const float4*)b2, n, 5, 1);
    // out2 = B [n x 128]

    // --- layer 3: 128 -> 64 (WMMA fp32), logF4 = 4 ---
    {
        dim3 grid(n / 16, 64 / 16);
        gemm_wmma_f32_kernel<128, 64><<<grid, 32, 0, stream>>>(B, W3, A);
    }
    scale4_kernel<<<cdiv(n << 4, TB), TB, 0, stream>>>((float4*)A, dinv, n, 4);
    hipMemsetAsync(B, 0, (size_t)n * 64 * sizeof(float), stream);
    scatter4_kernel<<<cdiv(E << 4, TB), TB, 0, stream>>>(src, dst, (const float4*)A, B, E, 4);
    combine4_kernel<<<cdiv(n << 4, TB), TB, 0, stream>>>((float4*)B, (const float4*)A, dinv,
                                                         (const float4*)b3, n, 4, 1);
    // out3 = B [n x 64]

    // --- layer 4: 64 -> 1 ---
    dot64_kernel<<<cdiv(n, TB), TB, 0, stream>>>(B, W4, A, n);
    scale1_kernel<<<cdiv(n, TB), TB, 0, stream>>>(A, dinv, n);
    hipMemsetAsync(B, 0, (size_t)n * sizeof(float), stream);
    scatter1_kernel<<<cdiv(E, TB), TB, 0, stream>>>(src, dst, A, B, E);
    combine_out_kernel<<<cdiv(n, TB), TB, 0, stream>>>(B, A, dinv, b4, (float*)d_out, n);
}